// PRGNN_72378788872337
// MI455X (gfx1250) — compile-verified
//
#include <hip/hip_runtime.h>

// ---------------------------------------------------------------------------
// ECC-GNN, refactored:  msg[e,o] = sum_d e_d * Z[src,d,o] + U[src,o]
// Per layer: pack B = [Wt(32x256) | Kb(32x32) | root(32x32)] -> Bp(32x320),
// then ONE branch-free WMMA f32 GEMM  P(M x 320) = X(M x 32) @ Bp(32 x 320).
// P cols: [0,256)=Z, [256,288)=U, [288,320)=R.
// ---------------------------------------------------------------------------

typedef float v2f __attribute__((ext_vector_type(2)));
typedef float v8f __attribute__((ext_vector_type(8)));

#define N_NODES 20000
#define N_EDGES 160000
#define N_PAIRS 50000
#define BCOLS   320
#define NT_N    (BCOLS/16)      // 20 column tiles
#define NT_M    (N_NODES/16)    // 1250 row tiles

// Gather knw/knb/root into contiguous Bp[k*320 + n]. Trivial 10240-thread kernel.
__global__ void __launch_bounds__(256)
pack_B(const float* __restrict__ knw,   // 8 x 1024  (d, i*32+o)
       const float* __restrict__ knb,   // 1024
       const float* __restrict__ root,  // 32 x 32
       float* __restrict__ Bp) {
    int i = blockIdx.x * blockDim.x + threadIdx.x;
    if (i >= 32 * BCOLS) return;
    int k = i / BCOLS, n = i % BCOLS;
    float v;
    if (n < 256)      v = knw[((n >> 5) << 10) + (k << 5) + (n & 31)];
    else if (n < 288) v = knb[(k << 5) + (n - 256)];
    else              v = root[(k << 5) + (n - 288)];
    Bp[i] = v;
}

// One 16x16 output tile per wave32; K=32 via 8x V_WMMA_F32_16X16X4_F32.
// Branch-free: packed B in, packed P out.
__global__ void __launch_bounds__(128)
ecc_gemm_wmma(const float* __restrict__ X,    // M x 32
              const float* __restrict__ Bp,   // 32 x 320
              float* __restrict__ P,          // M x 320
              int ntiles) {
    const int lane = threadIdx.x & 31;
    const int wave = threadIdx.x >> 5;
    const int tile = blockIdx.x * 4 + wave;
    if (tile >= ntiles) return;                 // wave-uniform guard
    const int mt = tile / NT_N;
    const int nt = tile % NT_N;
    const int m0 = mt << 4;
    const int n0 = nt << 4;

    // A 16x4 layout: lanes 0-15 row M=lane hold K0,K1; lanes 16-31 hold K2,K3.
    const int row   = lane & 15;
    const int khalf = (lane >> 4) << 1;         // 0 or 2
    const int col   = n0 + (lane & 15);

    const float* xp = X  + (m0 + row) * 32 + khalf;
    const float* bp = Bp + khalf * BCOLS + col;

    v8f acc = {};
#pragma unroll
    for (int kk = 0; kk < 8; ++kk) {
        v2f a, b;
        a.x = xp[kk * 4 + 0];                   // adjacent -> b64 load
        a.y = xp[kk * 4 + 1];
        b.x = bp[(kk * 4    ) * BCOLS];
        b.y = bp[(kk * 4 + 1) * BCOLS];
        acc = __builtin_amdgcn_wmma_f32_16x16x4_f32(
                  false, a, false, b, (short)0, acc, false, false);
    }

    // D layout: lanes 0-15 -> rows m0+0..7; lanes 16-31 -> rows m0+8..15.
    float* dst = P + (size_t)(m0 + ((lane >> 4) << 3)) * BCOLS + col;
#pragma unroll
    for (int v = 0; v < 8; ++v)
        dst[(size_t)v * BCOLS] = acc[v];
}

// 32 lanes per edge (lane == output channel o). P rows are L2-resident (25.6MB).
__global__ void __launch_bounds__(256)
edge_msg(const int*   __restrict__ edge_index,  // E x 2
         const float* __restrict__ e,           // E x 8
         const float* __restrict__ P,           // N x 320
         float*       __restrict__ agg,         // N x 32 (zeroed)
         int E) {
    int gid  = blockIdx.x * blockDim.x + threadIdx.x;
    int edge = gid >> 5;
    int o    = gid & 31;
    if (edge >= E) return;
    int src = edge_index[edge * 2 + 0];
    int tgt = edge_index[edge * 2 + 1];
    float ed[8];
#pragma unroll
    for (int d = 0; d < 8; ++d) ed[d] = e[edge * 8 + d];   // wave-uniform row
    const float* zr = P + (size_t)src * BCOLS + o;         // coalesced per d
    float m = zr[256];                                     // U term
#pragma unroll
    for (int d = 0; d < 8; ++d) m += ed[d] * zr[d * 32];
    atomicAdd(&agg[tgt * 32 + o], m);
}

__global__ void __launch_bounds__(256)
node_update(const float* __restrict__ agg, const float* __restrict__ P,
            const float* __restrict__ bias, float* __restrict__ h, int total) {
    int gid = blockIdx.x * blockDim.x + threadIdx.x;
    if (gid >= total) return;
    int n = gid >> 5, o = gid & 31;
    float v = agg[gid] + P[(size_t)n * BCOLS + 288 + o] + bias[o];
    h[gid] = v > 0.f ? v : 0.f;
}

__global__ void __launch_bounds__(256)
fill_zero(float* __restrict__ p, int n) {
    int i = blockIdx.x * blockDim.x + threadIdx.x;
    if (i < n) p[i] = 0.f;
}

// util[n] = h[n,:] . dw + db  (wave32 shuffle reduction, lane == channel)
__global__ void __launch_bounds__(256)
util_kernel(const float* __restrict__ h, const float* __restrict__ dw,
            const float* __restrict__ db, float* __restrict__ util, int N) {
    int gid = blockIdx.x * blockDim.x + threadIdx.x;
    int n = gid >> 5, o = gid & 31;
    if (n >= N) return;
    float v = h[n * 32 + o] * dw[o];
#pragma unroll
    for (int off = 16; off; off >>= 1) v += __shfl_xor(v, off, 32);
    if (o == 0) util[n] = v + db[0];
}

__global__ void __launch_bounds__(256)
pairs_kernel(const float* __restrict__ util, const int* __restrict__ ia,
             const int* __restrict__ ib, float* __restrict__ out, int P) {
    int p = blockIdx.x * blockDim.x + threadIdx.x;
    if (p < P) out[p] = util[ib[p]] - util[ia[p]];
}

extern "C" void kernel_launch(void* const* d_in, const int* in_sizes, int n_in,
                              void* d_out, int out_size, void* d_ws, size_t ws_size,
                              hipStream_t stream) {
    const float* x     = (const float*)d_in[0];
    const float* e     = (const float*)d_in[1];
    const float* knw1  = (const float*)d_in[2];
    const float* knb1  = (const float*)d_in[3];
    const float* root1 = (const float*)d_in[4];
    const float* bias1 = (const float*)d_in[5];
    const float* knw2  = (const float*)d_in[6];
    const float* knb2  = (const float*)d_in[7];
    const float* root2 = (const float*)d_in[8];
    const float* bias2 = (const float*)d_in[9];
    const float* dw    = (const float*)d_in[10];
    const float* db    = (const float*)d_in[11];
    const int*   eidx  = (const int*)d_in[12];
    const int*   idx_a = (const int*)d_in[13];
    const int*   idx_b = (const int*)d_in[14];
    float* out = (float*)d_out;

    char* ws = (char*)d_ws;
    float* Bp   = (float*)(ws);                  //     40,960 B (pad to 64K)
    float* Pm   = (float*)(ws + 65536);          // 25,600,000 B
    float* agg  = (float*)(ws + 25665536);       //  2,560,000 B
    float* h1   = (float*)(ws + 28225536);       //  2,560,000 B
    float* h2   = (float*)(ws + 30785536);       //  2,560,000 B
    float* util = (float*)(ws + 33345536);       //     80,000 B

    const int ntiles    = NT_M * NT_N;           // 25000 tiles, 4 waves/block
    const int gemm_grid = (ntiles + 3) / 4;
    const int nf        = N_NODES * 32;          // 640000
    const int elem_grid = (nf + 255) / 256;
    const int edge_grid = (N_EDGES * 32 + 255) / 256;
    const int pair_grid = (N_PAIRS + 255) / 256;
    const int pack_grid = (32 * BCOLS + 255) / 256;

    // -------- layer 1 --------
    fill_zero<<<elem_grid, 256, 0, stream>>>(agg, nf);
    pack_B<<<pack_grid, 256, 0, stream>>>(knw1, knb1, root1, Bp);
    ecc_gemm_wmma<<<gemm_grid, 128, 0, stream>>>(x, Bp, Pm, ntiles);
    edge_msg<<<edge_grid, 256, 0, stream>>>(eidx, e, Pm, agg, N_EDGES);
    node_update<<<elem_grid, 256, 0, stream>>>(agg, Pm, bias1, h1, nf);

    // -------- layer 2 --------
    fill_zero<<<elem_grid, 256, 0, stream>>>(agg, nf);
    pack_B<<<pack_grid, 256, 0, stream>>>(knw2, knb2, root2, Bp);
    ecc_gemm_wmma<<<gemm_grid, 128, 0, stream>>>(h1, Bp, Pm, ntiles);
    edge_msg<<<edge_grid, 256, 0, stream>>>(eidx, e, Pm, agg, N_EDGES);
    node_update<<<elem_grid, 256, 0, stream>>>(agg, Pm, bias2, h2, nf);

    // -------- readout --------
    util_kernel<<<elem_grid, 256, 0, stream>>>(h2, dw, db, util, N_NODES);
    pairs_kernel<<<pair_grid, 256, 0, stream>>>(util, idx_a, idx_b, out, N_PAIRS);
}